// OnlineTripletLoss_33827162423929
// MI455X (gfx1250) — compile-verified
//
#include <hip/hip_runtime.h>

#define NN 4096
#define DD 512
#define MARGIN 1.0f
#define FINF 1e30f

// --- LDS tiling config -----------------------------------------------------
#define KC    32                    // K-chunk (floats)
#define RS    36                    // LDS row stride = KC + 4 pad DWORDs (TDM pad)
#define AROWS 128                   // i-rows per block tile
#define BROWS 64                    // j-cols per block tile
#define NCH   (DD / KC)             // 16 K-chunks
#define A_BUF (AROWS * RS)          // floats per A buffer
#define B_BUF (BROWS * RS)          // floats per B buffer
#define SMEM_FLOATS (2 * A_BUF + 2 * B_BUF)
#define SMEM_BYTES  (SMEM_FLOATS * 4)   // 55296 B

typedef __attribute__((ext_vector_type(2))) float        v2f;
typedef __attribute__((ext_vector_type(8))) float        v8f;
typedef __attribute__((ext_vector_type(4))) unsigned int v4u;
typedef __attribute__((ext_vector_type(8))) int          v8i;
typedef __attribute__((ext_vector_type(4))) int          v4i;

// Low 32 bits of the flat address of a __shared__ object == LDS byte offset
// (ISA: LDS aperture keeps the offset in addr[31:0]).
__device__ __forceinline__ unsigned lds_off(const void* p) {
  return (unsigned)(unsigned long long)(uintptr_t)p;
}

// Read 8 bytes from LDS at a raw byte offset through an addrspace(3) pointer:
// guarantees ds_load_b64 (no flat path).
__device__ __forceinline__ v2f lds_v2f(unsigned byteOff) {
  typedef __attribute__((address_space(3))) const v2f* lds_p;
  return *reinterpret_cast<lds_p>(byteOff);
}

// ---------------------------------------------------------------------------
// TDM: DMA a rows x KC fp32 tile (row stride DD) from global into LDS,
// padding 4 DWORDs every 32 DWORDs (-> LDS row stride RS=36, bank-conflict
// free for the WMMA ds_load pattern).  D# per CDNA5 ISA ch.8.
// ---------------------------------------------------------------------------
__device__ __forceinline__ void tdm_tile_to_lds(const float* tile, unsigned lds,
                                                unsigned rows) {
  unsigned long long ga = (unsigned long long)(uintptr_t)tile;
  v4u g0;
  g0[0] = 1u;                                   // count=1 (valid), user mode
  g0[1] = lds;                                  // lds_addr (bytes)
  g0[2] = (unsigned)ga;                         // global_addr[31:0]
  g0[3] = (unsigned)((ga >> 32) & 0x01FFFFFFu)  // global_addr[56:32]
        | (2u << 30);                           // type=2 ("image")
  v8i g1;
  g1[0] = (int)((2u << 16)      // data_size = 4B
              | (1u << 20)      // pad_enable
              | (4u << 22)      // pad_interval code 4 = 32 DWORDs
              | (3u << 25));    // pad_amount  code 3 = 4 DWORDs
  g1[1] = (int)(((unsigned)DD & 0xFFFFu) << 16);                    // tensor_dim0 lo
  g1[2] = (int)((((unsigned)DD >> 16) & 0xFFFFu)
              | (((unsigned)NN & 0xFFFFu) << 16));                  // dim0 hi | dim1 lo
  g1[3] = (int)((((unsigned)NN >> 16) & 0xFFFFu)
              | ((unsigned)KC << 16));                              // dim1 hi | tile_dim0
  g1[4] = (int)rows;                                                // tile_dim1 (tile_dim2=0)
  g1[5] = DD;                                                       // tensor_dim0_stride lo
  g1[6] = 0;
  g1[7] = 0;
  v4i z4 = {0, 0, 0, 0};                        // 2D tensor: groups 2/3 unused
  v8i z8 = {0, 0, 0, 0, 0, 0, 0, 0};
  __builtin_amdgcn_tensor_load_to_lds(g0, g1, z4, z4, z8, 0);
}

// ---------------------------------------------------------------------------
// Kernel 0: per-row squared norms (one wave per row) + workspace init
// ---------------------------------------------------------------------------
__global__ void k_rownorms(const float* __restrict__ E,
                           float* __restrict__ sq_norms,
                           float* __restrict__ min_sq,
                           float* __restrict__ loss_sum,
                           unsigned* __restrict__ cnt,
                           unsigned* __restrict__ correct) {
  const int lane = threadIdx.x & 31;
  const int row  = (blockIdx.x * (blockDim.x >> 5)) + (threadIdx.x >> 5);
  if (blockIdx.x == 0 && threadIdx.x == 0) {
    *loss_sum = 0.0f;
    *cnt = 0u;
    *correct = 0u;
  }
  if (row >= NN) return;
  const float* r = E + (size_t)row * DD;
  float s = 0.0f;
#pragma unroll
  for (int k = 0; k < DD; k += 32 * 4) {
    float4 v = *(const float4*)(r + k + lane * 4);
    s += v.x * v.x + v.y * v.y + v.z * v.z + v.w * v.w;
  }
#pragma unroll
  for (int m = 16; m >= 1; m >>= 1) s += __shfl_xor(s, m, 32);
  if (lane == 0) {
    sq_norms[row] = s;
    min_sq[row]   = FINF;   // positive float: int-ordered atomicMin valid
  }
}

// ---------------------------------------------------------------------------
// One K-chunk of WMMA from LDS via raw-offset ds_load_b64.
// aByte/bByte already include the per-lane term ((row)*RS + 2*half)*4.
// ---------------------------------------------------------------------------
__device__ __forceinline__ void gemm_chunk(unsigned aByte, unsigned bByte,
                                           v8f c[4]) {
#pragma unroll
  for (int k = 0; k < KC; k += 4) {
    v2f av = lds_v2f(aByte + k * 4u);
    v2f b0 = lds_v2f(bByte + k * 4u);
    v2f b1 = lds_v2f(bByte + 16u * RS * 4u + k * 4u);
    v2f b2 = lds_v2f(bByte + 32u * RS * 4u + k * 4u);
    v2f b3 = lds_v2f(bByte + 48u * RS * 4u + k * 4u);
    c[0] = __builtin_amdgcn_wmma_f32_16x16x4_f32(false, av, false, b0, (short)0, c[0], false, false);
    c[1] = __builtin_amdgcn_wmma_f32_16x16x4_f32(false, av, false, b1, (short)0, c[1], false, false);
    c[2] = __builtin_amdgcn_wmma_f32_16x16x4_f32(false, av, false, b2, (short)0, c[2], false, false);
    c[3] = __builtin_amdgcn_wmma_f32_16x16x4_f32(false, av, false, b3, (short)0, c[3], false, false);
  }
}

// ---------------------------------------------------------------------------
// Block GEMM: 128x64 tile of G = E*E^T, TDM-staged, double-buffered LDS.
// Wave w owns rows [ib*128+16w, +16), all 64 cols (4 accumulators).
// ---------------------------------------------------------------------------
__device__ __forceinline__ void block_gemm(const float* __restrict__ E,
                                           int ib, int jb,
                                           int w, int lm, int half,
                                           unsigned ldsBase, v8f c[4]) {
  const unsigned aOff0 = ldsBase;
  const unsigned aOff1 = ldsBase + (unsigned)A_BUF * 4u;
  const unsigned bOff0 = ldsBase + 2u * (unsigned)A_BUF * 4u;
  const unsigned bOff1 = bOff0 + (unsigned)B_BUF * 4u;
  const unsigned aLane = (unsigned)((16 * w + lm) * RS + 2 * half) * 4u;
  const unsigned bLane = (unsigned)(lm * RS + 2 * half) * 4u;

  const bool ld = (threadIdx.x < 32);          // wave 0 drives the TDM
  const float* Abase = E + (size_t)(ib * AROWS) * DD;
  const float* Bbase = E + (size_t)(jb * BROWS) * DD;

  if (ld) {
    tdm_tile_to_lds(Abase, aOff0, AROWS);
    tdm_tile_to_lds(Bbase, bOff0, BROWS);
    __builtin_amdgcn_s_wait_tensorcnt(0);
  }
  __syncthreads();

  for (int ch = 0; ch < NCH; ++ch) {
    const int cur = ch & 1;
    if (ld && (ch + 1 < NCH)) {                // prefetch next chunk via TDM
      tdm_tile_to_lds(Abase + (ch + 1) * KC, cur ? aOff0 : aOff1, AROWS);
      tdm_tile_to_lds(Bbase + (ch + 1) * KC, cur ? bOff0 : bOff1, BROWS);
    }
    gemm_chunk((cur ? aOff1 : aOff0) + aLane, (cur ? bOff1 : bOff0) + bLane, c);
    if (ld && (ch + 1 < NCH)) __builtin_amdgcn_s_wait_tensorcnt(0);
    __syncthreads();
  }
}

// ---------------------------------------------------------------------------
// Pass A: hardest-negative mining: min_sq[i] = min masked sq_dist row-min
// ---------------------------------------------------------------------------
__global__ void __launch_bounds__(256) k_minneg(const float* __restrict__ E,
                                                const int* __restrict__ tgt,
                                                const float* __restrict__ sq_norms,
                                                float* __restrict__ min_sq) {
  extern __shared__ float smem[];
  const int lane = threadIdx.x & 31;
  const int lm   = lane & 15;
  const int half = lane >> 4;
  const int w    = threadIdx.x >> 5;
  const int ib   = blockIdx.x >> 6;            // 32 i-blocks of 128
  const int jb   = blockIdx.x & 63;            // 64 j-blocks of 64

  v8f c[4] = {v8f{}, v8f{}, v8f{}, v8f{}};
  block_gemm(E, ib, jb, w, lm, half, lds_off(smem), c);

  const int i16 = ib * AROWS + w * 16;
  const int j64 = jb * BROWS;
  int   jl[4];
  float jn[4];
#pragma unroll
  for (int q = 0; q < 4; ++q) {
    int j = j64 + q * 16 + lm;
    jl[q] = tgt[j];
    jn[q] = sq_norms[j];
  }
#pragma unroll
  for (int r = 0; r < 8; ++r) {
    const int   i  = i16 + r + half * 8;       // wave32 C layout: half1 => M=r+8
    const int   il = tgt[i];
    const float in = sq_norms[i];
    float v = FINF;
#pragma unroll
    for (int q = 0; q < 4; ++q) {
      float sq = fmaxf(in + jn[q] - 2.0f * c[q][r], 0.0f);
      v = fminf(v, (jl[q] == il) ? FINF : sq);
    }
#pragma unroll
    for (int m = 8; m >= 1; m >>= 1) v = fminf(v, __shfl_xor(v, m, 32));
    if (lm == 0) atomicMin((int*)&min_sq[i], __float_as_int(v));
  }
}

// ---------------------------------------------------------------------------
// Pass B: recompute tiles (upper triangle) and accumulate loss / cnt / correct
// ---------------------------------------------------------------------------
__global__ void __launch_bounds__(256) k_loss(const float* __restrict__ E,
                                              const int* __restrict__ tgt,
                                              const float* __restrict__ sq_norms,
                                              const float* __restrict__ min_sq,
                                              float* __restrict__ loss_sum,
                                              unsigned* __restrict__ cnt,
                                              unsigned* __restrict__ correct) {
  extern __shared__ float smem[];
  const int lane = threadIdx.x & 31;
  const int lm   = lane & 15;
  const int half = lane >> 4;
  const int w    = threadIdx.x >> 5;
  const int ib   = blockIdx.x >> 6;
  const int jb   = blockIdx.x & 63;
  if (jb * BROWS + (BROWS - 1) <= ib * AROWS) return;  // block-uniform: no j > i

  v8f c[4] = {v8f{}, v8f{}, v8f{}, v8f{}};
  block_gemm(E, ib, jb, w, lm, half, lds_off(smem), c);

  const int i16 = ib * AROWS + w * 16;
  const int j64 = jb * BROWS;
  int   jl[4];
  float jn[4];
#pragma unroll
  for (int q = 0; q < 4; ++q) {
    int j = j64 + q * 16 + lm;
    jl[q] = tgt[j];
    jn[q] = sq_norms[j];
  }

  float    lsum = 0.0f;
  unsigned lcnt = 0u, lcor = 0u;
#pragma unroll
  for (int r = 0; r < 8; ++r) {
    const int   i   = i16 + r + half * 8;
    const int   il  = tgt[i];
    const float in  = sq_norms[i];
    const float msq = min_sq[i];
    const float mn  = sqrtf(msq);
#pragma unroll
    for (int q = 0; q < 4; ++q) {
      const int j = j64 + q * 16 + lm;
      if (j > i && jl[q] == il) {
        float sq = fmaxf(in + jn[q] - 2.0f * c[q][r], 0.0f);
        float d  = sqrtf(sq);
        if (d - mn + MARGIN > 0.0f) {             // mined validity (euclidean)
          lcnt += 1u;
          lsum += fmaxf(sq - msq + MARGIN, 0.0f); // squared-distance loss
          lcor += (sq < msq) ? 1u : 0u;
        }
      }
    }
  }
#pragma unroll
  for (int m = 16; m >= 1; m >>= 1) {
    lsum += __shfl_xor(lsum, m, 32);
    lcnt += __shfl_xor(lcnt, m, 32);
    lcor += __shfl_xor(lcor, m, 32);
  }
  if (lane == 0) {
    atomicAdd(loss_sum, lsum);
    atomicAdd(cnt, lcnt);
    atomicAdd(correct, lcor);
  }
}

// ---------------------------------------------------------------------------
// Finalize: loss = sum/max(cnt,1), accuracy = correct/max(cnt,1)
// ---------------------------------------------------------------------------
__global__ void k_final(const float* __restrict__ loss_sum,
                        const unsigned* __restrict__ cnt,
                        const unsigned* __restrict__ correct,
                        float* __restrict__ out) {
  unsigned c = *cnt;
  float denom = (float)(c > 0u ? c : 1u);
  out[0] = *loss_sum / denom;
  out[1] = (float)(*correct) / denom;
}

extern "C" void kernel_launch(void* const* d_in, const int* in_sizes, int n_in,
                              void* d_out, int out_size, void* d_ws, size_t ws_size,
                              hipStream_t stream) {
  const float* E   = (const float*)d_in[0];
  const int*   tgt = (const int*)d_in[1];
  float*       out = (float*)d_out;

  float*    sq_norms = (float*)d_ws;
  float*    min_sq   = sq_norms + NN;
  float*    loss_sum = min_sq + NN;
  unsigned* cnt      = (unsigned*)(loss_sum + 1);
  unsigned* correct  = cnt + 1;

  // 4096 waves (one per row), 8 waves per 256-thread block
  k_rownorms<<<NN / 8, 256, 0, stream>>>(E, sq_norms, min_sq, loss_sum, cnt, correct);

  // 32 i-blocks x 64 j-blocks, 256 threads, 54 KB LDS (TDM double buffers)
  k_minneg<<<32 * 64, 256, SMEM_BYTES, stream>>>(E, tgt, sq_norms, min_sq);
  k_loss<<<32 * 64, 256, SMEM_BYTES, stream>>>(E, tgt, sq_norms, min_sq,
                                               loss_sum, cnt, correct);

  k_final<<<1, 1, 0, stream>>>(loss_sum, cnt, correct, out);
}